// SpatialAttention_44770739094057
// MI455X (gfx1250) — compile-verified
//
#include <hip/hip_runtime.h>
#include <hip/hip_bf16.h>
#include <cstdint>

typedef __attribute__((ext_vector_type(2))) float v2f;
typedef __attribute__((ext_vector_type(8))) float v8f;

#define D 128
#define H 8
#define HD 16
#define ED 32
#define LDS_PITCH 132   // 128 dwords + TDM pad of 4 dwords -> conflict-free column reads

#if defined(__AMDGCN__) && __has_builtin(__builtin_amdgcn_tensor_load_to_lds) && \
    __has_builtin(__builtin_amdgcn_s_wait_tensorcnt)
#define USE_TDM 1
#else
#define USE_TDM 0
#endif

#if USE_TDM
typedef __attribute__((ext_vector_type(4))) unsigned tdm_v4u;
typedef __attribute__((ext_vector_type(8))) int      tdm_v8i;
typedef __attribute__((ext_vector_type(4))) int      tdm_v4i;

// DMA a 16x128 fp32 tile (row stride 128 floats in memory) into LDS with a
// 4-dword pad every 128 dwords (LDS row pitch = 132 dwords).
__device__ __forceinline__ void tdm_load_tile_16x128(const float* gsrc, void* lds_dst) {
    const unsigned long long ga = (unsigned long long)(uintptr_t)gsrc;
    const unsigned lds = (unsigned)(uintptr_t)lds_dst;   // low 32 bits = LDS offset
    tdm_v4u g0;
    g0[0] = 1u;                                          // count=1, user mode
    g0[1] = lds;                                         // lds_addr
    g0[2] = (unsigned)ga;                                // global_addr[31:0]
    g0[3] = (unsigned)((ga >> 32) & 0x01ffffffu) | (2u << 30);  // addr[56:32] | type=2
    tdm_v8i g1;
    // data_size=4B(2), pad_enable, pad_interval=128dw(6), pad_amount=4dw(3)
    g1[0] = (int)((2u << 16) | (1u << 20) | (6u << 22) | (3u << 25));
    g1[1] = (int)(128u << 16);           // tensor_dim0 = 128 (bits 79:48, low half)
    g1[2] = (int)(16u << 16);            // tensor_dim0 hi=0 | tensor_dim1 = 16
    g1[3] = (int)(128u << 16);           // tensor_dim1 hi=0 | tile_dim0 = 128
    g1[4] = (int)(16u);                  // tile_dim1 = 16 | tile_dim2 = 0
    g1[5] = (int)(128);                  // tensor_dim0_stride = 128
    g1[6] = 0;
    g1[7] = 0;
    tdm_v4i gz = {0, 0, 0, 0};
#if __clang_major__ >= 23
    tdm_v8i gz8 = {0, 0, 0, 0, 0, 0, 0, 0};
    __builtin_amdgcn_tensor_load_to_lds(g0, g1, gz, gz, gz8, 0);
#else
    __builtin_amdgcn_tensor_load_to_lds(g0, g1, gz, gz, 0);
#endif
}
#endif

// ---- ordered-uint encoding for deterministic float atomic max ----
__device__ __forceinline__ unsigned f2ord(float f) {
    unsigned u = __float_as_uint(f);
    return (u & 0x80000000u) ? ~u : (u | 0x80000000u);
}
__device__ __forceinline__ float ord2f(unsigned u) {
    return (u & 0x80000000u) ? __uint_as_float(u & 0x7fffffffu)
                             : __uint_as_float(~u);
}

// ---------------------------------------------------------------
// init workspace: agg=0, ssum=0, max=ordered(-inf)=0
// ---------------------------------------------------------------
__global__ void init_ws(float* __restrict__ agg, unsigned* __restrict__ m,
                        float* __restrict__ ssum, int n128, int n8) {
    int i = blockIdx.x * blockDim.x + threadIdx.x;
    if (i < n128) agg[i] = 0.0f;
    if (i < n8) { m[i] = 0u; ssum[i] = 0.0f; }
}

// ---------------------------------------------------------------
// q/k/v projection. One 768-thread block (24 waves) per 16-row tile:
// TDM stages the 16x128 x-tile into LDS once; wave w computes the
// 16x16 tile for matrix (w>>3) and columns ((w&7)*16) via fp32 WMMA.
// ---------------------------------------------------------------
__global__ void __launch_bounds__(768) qkv_wmma(
        const float* __restrict__ x,
        const float* __restrict__ Wq, const float* __restrict__ bq,
        const float* __restrict__ Wk, const float* __restrict__ bk,
        const float* __restrict__ Wv, const float* __restrict__ bv,
        float* __restrict__ q, float* __restrict__ k, float* __restrict__ v) {
    __shared__ float xt[16][LDS_PITCH];
    const int row0 = blockIdx.x * 16;

#if USE_TDM
    if (threadIdx.x < 32) {
        tdm_load_tile_16x128(x + (size_t)row0 * D, &xt[0][0]);
        __builtin_amdgcn_s_wait_tensorcnt(0);
    }
#else
    for (int i = threadIdx.x; i < 16 * D; i += 768)
        xt[i >> 7][i & 127] = x[(size_t)row0 * D + i];
#endif
    __syncthreads();

    const int w    = threadIdx.x >> 5;    // 0..23
    const int mat  = w >> 3;              // 0=q 1=k 2=v
    const int col0 = (w & 7) * 16;
    const float* W    = (mat == 0) ? Wq : (mat == 1) ? Wk : Wv;
    const float* bias = (mat == 0) ? bq : (mat == 1) ? bk : bv;
    float*       out  = (mat == 0) ? q  : (mat == 1) ? k  : v;

    const int lane = threadIdx.x & 31;
    const int grp  = lane >> 4;           // K-pair group
    const int idx  = lane & 15;           // M (A) / N (B, C)

    v8f c = {};
    for (int kb = 0; kb < D; kb += 4) {
        v2f a, b;
        a.x = xt[idx][kb + 2 * grp + 0];
        a.y = xt[idx][kb + 2 * grp + 1];
        b.x = W[(size_t)(kb + 2 * grp + 0) * D + col0 + idx];
        b.y = W[(size_t)(kb + 2 * grp + 1) * D + col0 + idx];
        c = __builtin_amdgcn_wmma_f32_16x16x4_f32(false, a, false, b,
                                                  (short)0, c, false, false);
    }
    const float bb = bias[col0 + idx];
    for (int r = 0; r < 8; ++r) {
        out[(size_t)(row0 + r + 8 * grp) * D + col0 + idx] = c[r] + bb;
    }
}

// ---------------------------------------------------------------
// per-edge attention score + edge bias + per-dst running max
// ---------------------------------------------------------------
__global__ void edge_score(const int* __restrict__ ei,
                           const float* __restrict__ ea,
                           const float* __restrict__ q,
                           const float* __restrict__ k,
                           const float* __restrict__ We,
                           const float* __restrict__ be,
                           float* __restrict__ score,
                           unsigned* __restrict__ m, int E) {
    int e = blockIdx.x * blockDim.x + threadIdx.x;
    if (e >= E) return;
    const int s = ei[e];
    const int d = ei[E + e];
    const float4* qd = (const float4*)(q + (size_t)d * D);
    const float4* ks = (const float4*)(k + (size_t)s * D);

    float bias[H];
#pragma unroll
    for (int h = 0; h < H; ++h) bias[h] = be[h];
    for (int j = 0; j < ED; ++j) {
        const float a = ea[(size_t)e * ED + j];
#pragma unroll
        for (int h = 0; h < H; ++h) bias[h] += a * We[j * H + h];
    }

#pragma unroll
    for (int h = 0; h < H; ++h) {
        float acc = 0.0f;
#pragma unroll
        for (int t = 0; t < 4; ++t) {
            const float4 a = qd[h * 4 + t];
            const float4 b = ks[h * 4 + t];
            acc += a.x * b.x + a.y * b.y + a.z * b.z + a.w * b.w;
        }
        const float sc = acc * 0.25f + bias[h];   // 1/sqrt(HD=16)
        score[(size_t)e * H + h] = sc;
        atomicMax(&m[(size_t)d * H + h], f2ord(sc));
    }
}

// ---------------------------------------------------------------
// exp(score - max[dst]) and per-dst sum
// ---------------------------------------------------------------
__global__ void edge_exp(const int* __restrict__ ei,
                         float* __restrict__ score,
                         const unsigned* __restrict__ m,
                         float* __restrict__ ssum, int E) {
    int e = blockIdx.x * blockDim.x + threadIdx.x;
    if (e >= E) return;
    const int d = ei[E + e];
#pragma unroll
    for (int h = 0; h < H; ++h) {
        const float sc = score[(size_t)e * H + h];
        const float mx = ord2f(m[(size_t)d * H + h]);
        const float ex = expf(sc - mx);
        score[(size_t)e * H + h] = ex;
        atomicAdd(&ssum[(size_t)d * H + h], ex);
    }
}

// ---------------------------------------------------------------
// weighted message scatter-add: agg[dst] += attn * v[src]
// ---------------------------------------------------------------
__global__ void edge_scatter(const int* __restrict__ ei,
                             const float* __restrict__ score,
                             const float* __restrict__ ssum,
                             const float* __restrict__ v,
                             float* __restrict__ agg, int E) {
    int e = blockIdx.x * blockDim.x + threadIdx.x;
    if (e >= E) return;
    const int s = ei[e];
    const int d = ei[E + e];
    const float4* vs = (const float4*)(v + (size_t)s * D);
    float* out = agg + (size_t)d * D;
#pragma unroll
    for (int h = 0; h < H; ++h) {
        const float a = score[(size_t)e * H + h] /
                        (ssum[(size_t)d * H + h] + 1e-8f);
#pragma unroll
        for (int t = 0; t < 4; ++t) {
            const float4 vv = vs[h * 4 + t];
            const int base = h * HD + t * 4;
            atomicAdd(&out[base + 0], a * vv.x);
            atomicAdd(&out[base + 1], a * vv.y);
            atomicAdd(&out[base + 2], a * vv.z);
            atomicAdd(&out[base + 3], a * vv.w);
        }
    }
}

// ---------------------------------------------------------------
// out = LN(agg @ Wo + bo + x). 256 threads per 16-row tile: TDM
// stages the agg tile, 8 waves each do one 16x16 fp32 WMMA tile,
// then block-cooperative LayerNorm in LDS (eps = 1e-5).
// ---------------------------------------------------------------
__global__ void __launch_bounds__(256) out_ln(
        const float* __restrict__ agg,
        const float* __restrict__ Wo, const float* __restrict__ bo,
        const float* __restrict__ x,
        const float* __restrict__ gamma, const float* __restrict__ beta,
        float* __restrict__ out) {
    __shared__ float at[16][LDS_PITCH];    // staged agg tile (TDM)
    __shared__ float tile[16][LDS_PITCH];  // projected + residual result
    __shared__ float redS[16][17];
    __shared__ float redQ[16][17];
    __shared__ float muS[16];
    __shared__ float rsS[16];

    const int row0 = blockIdx.x * 16;

#if USE_TDM
    if (threadIdx.x < 32) {
        tdm_load_tile_16x128(agg + (size_t)row0 * D, &at[0][0]);
        __builtin_amdgcn_s_wait_tensorcnt(0);
    }
#else
    for (int i = threadIdx.x; i < 16 * D; i += 256)
        at[i >> 7][i & 127] = agg[(size_t)row0 * D + i];
#endif
    __syncthreads();

    const int wave = threadIdx.x >> 5;   // 0..7 -> col tile
    const int lane = threadIdx.x & 31;
    const int grp  = lane >> 4;
    const int idx  = lane & 15;
    const int col0 = wave * 16;

    v8f c = {};
    for (int kb = 0; kb < D; kb += 4) {
        v2f a, b;
        a.x = at[idx][kb + 2 * grp + 0];
        a.y = at[idx][kb + 2 * grp + 1];
        b.x = Wo[(size_t)(kb + 2 * grp + 0) * D + col0 + idx];
        b.y = Wo[(size_t)(kb + 2 * grp + 1) * D + col0 + idx];
        c = __builtin_amdgcn_wmma_f32_16x16x4_f32(false, a, false, b,
                                                  (short)0, c, false, false);
    }
    const int col = col0 + idx;
    const float bb = bo[col];
    for (int r = 0; r < 8; ++r) {
        const int row = r + 8 * grp;
        tile[row][col] = c[r] + bb + x[(size_t)(row0 + row) * D + col];
    }
    __syncthreads();

    const int rrow = threadIdx.x >> 4;
    const int sub  = threadIdx.x & 15;
    float s = 0.0f, sq = 0.0f;
#pragma unroll
    for (int j = 0; j < 8; ++j) {
        const float t = tile[rrow][sub * 8 + j];
        s += t; sq += t * t;
    }
    redS[rrow][sub] = s;
    redQ[rrow][sub] = sq;
    __syncthreads();
    if (threadIdx.x < 16) {
        float ts = 0.0f, tq = 0.0f;
#pragma unroll
        for (int j = 0; j < 16; ++j) { ts += redS[threadIdx.x][j]; tq += redQ[threadIdx.x][j]; }
        const float mu  = ts * (1.0f / 128.0f);
        const float var = tq * (1.0f / 128.0f) - mu * mu;
        muS[threadIdx.x] = mu;
        rsS[threadIdx.x] = rsqrtf(var + 1e-5f);
    }
    __syncthreads();
    const float mu = muS[rrow];
    const float rs = rsS[rrow];
#pragma unroll
    for (int j = 0; j < 8; ++j) {
        const int cc = sub * 8 + j;
        out[(size_t)(row0 + rrow) * D + cc] =
            (tile[rrow][cc] - mu) * rs * gamma[cc] + beta[cc];
    }
}

extern "C" void kernel_launch(void* const* d_in, const int* in_sizes, int n_in,
                              void* d_out, int out_size, void* d_ws, size_t ws_size,
                              hipStream_t stream) {
    (void)n_in; (void)out_size; (void)ws_size;
    const float* x     = (const float*)d_in[0];
    const int*   ei    = (const int*)d_in[1];
    const float* ea    = (const float*)d_in[2];
    const float* Wq    = (const float*)d_in[3];
    const float* bq    = (const float*)d_in[4];
    const float* Wk    = (const float*)d_in[5];
    const float* bk    = (const float*)d_in[6];
    const float* Wv    = (const float*)d_in[7];
    const float* bv    = (const float*)d_in[8];
    const float* We    = (const float*)d_in[9];
    const float* be    = (const float*)d_in[10];
    const float* Wo    = (const float*)d_in[11];
    const float* bo    = (const float*)d_in[12];
    const float* gamma = (const float*)d_in[13];
    const float* beta  = (const float*)d_in[14];
    float* out = (float*)d_out;

    const int N = in_sizes[0] / D;    // 50000
    const int E = in_sizes[2] / ED;   // 800000
    const size_t qN = (size_t)N * D;

    float* ws    = (float*)d_ws;
    float* q     = ws;
    float* k     = ws + qN;
    float* v     = ws + 2 * qN;
    float* agg   = ws + 3 * qN;
    float* score = ws + 4 * qN;                               // E*H
    unsigned* m  = (unsigned*)(ws + 4 * qN + (size_t)E * H);  // N*H
    float* ssum  = ws + 4 * qN + (size_t)E * H + (size_t)N * H;

    const int n128 = N * D;
    init_ws<<<(n128 + 255) / 256, 256, 0, stream>>>(agg, m, ssum, n128, N * H);

    qkv_wmma<<<(N + 15) / 16, 768, 0, stream>>>(x, Wq, bq, Wk, bk, Wv, bv, q, k, v);

    const int eb = (E + 255) / 256;
    edge_score<<<eb, 256, 0, stream>>>(ei, ea, q, k, We, be, score, m, E);
    edge_exp<<<eb, 256, 0, stream>>>(ei, score, m, ssum, E);
    edge_scatter<<<eb, 256, 0, stream>>>(ei, score, ssum, v, agg, E);

    out_ln<<<(N + 15) / 16, 256, 0, stream>>>(agg, Wo, bo, x, gamma, beta, out);
}